// STABR_48550310314246
// MI455X (gfx1250) — compile-verified
//
#include <hip/hip_runtime.h>
#include <math.h>

// ---- problem constants (match reference) ----
constexpr int kB = 256, kT = 100, kK = 5;
constexpr int kU = 64, kSE = 50, kTE = 25, kPE = 10, kVL = 50, kSV = 100000;
constexpr int k3U = 3 * kU; // 192

// ---- workspace layout (in floats) ----
constexpr size_t A_STRIDE   = (size_t)kB * kT * k3U;      // one precomputed "a" array
constexpr size_t GI_OFF     = 4 * A_STRIDE;               // gi: (B,T,TE)
constexpr size_t SEQ_OFF    = GI_OFF + (size_t)kB * kT * kTE;
constexpr size_t SEQ_STRIDE = (size_t)kB * kT * 128;      // per path (song/tag)
constexpr size_t H_OFF      = SEQ_OFF + 2 * SEQ_STRIDE;   // forward final hidden per path
constexpr size_t CTX_OFF    = H_OFF + 2 * (size_t)kB * kU;
constexpr size_t V_OFF      = CTX_OFF + (size_t)kB * 256; // padded v: (B,64)

typedef __attribute__((ext_vector_type(2))) float v2f;
typedef __attribute__((ext_vector_type(8))) float v8f;

__device__ __forceinline__ float hsig(float x) {
    return fminf(fmaxf(0.2f * x + 0.5f, 0.f), 1.f);
}

// ---------------------------------------------------------------------------
// gi = masked mean of tag embeddings over K
// ---------------------------------------------------------------------------
__global__ void gi_kernel(const int* __restrict__ tags,
                          const float* __restrict__ tag_emb,
                          float* __restrict__ ws) {
    int idx = blockIdx.x * blockDim.x + threadIdx.x; // over B*T
    if (idx >= kB * kT) return;
    const int* tg = tags + (size_t)idx * kK;
    float den = 0.f;
    for (int k = 0; k < kK; k++) den += (tg[k] != 0) ? 1.f : 0.f;
    float d = (den == 0.f) ? 1.f : den;
    float* gi = ws + GI_OFF + (size_t)idx * kTE;
    for (int e = 0; e < kTE; e++) {
        float s = 0.f;
        for (int k = 0; k < kK; k++)
            if (tg[k] != 0) s += tag_emb[(size_t)tg[k] * kTE + e];
        gi[e] = s / d;
    }
}

// ---------------------------------------------------------------------------
// a = x @ kernel + skip @ skipk + ib + sb, for 4 (path,dir) combos
// which: 0=s_f 1=s_b 2=t_f 3=t_b ; block = 192 threads, one block per (b*t)
// ---------------------------------------------------------------------------
__global__ void precompute_a_kernel(const int* __restrict__ songs,
                                    const int* __restrict__ skips,
                                    const float* __restrict__ song_emb,
                                    const float* __restrict__ skip_emb,
                                    const float* __restrict__ sk_f, const float* __restrict__ ss_f, const float* __restrict__ sb_f,
                                    const float* __restrict__ sk_b, const float* __restrict__ ss_b, const float* __restrict__ sb_b,
                                    const float* __restrict__ tk_f, const float* __restrict__ ts_f, const float* __restrict__ tb_f,
                                    const float* __restrict__ tk_b, const float* __restrict__ ts_b, const float* __restrict__ tb_b,
                                    float* __restrict__ ws) {
    int bt = blockIdx.x;
    int which = blockIdx.y;
    int j = threadIdx.x; // 0..191

    __shared__ float xv[kSE];
    __shared__ float sv[kPE];

    const float* kern; const float* skk; const float* bias; int xdim;
    if (which == 0)      { kern = sk_f; skk = ss_f; bias = sb_f; xdim = kSE; }
    else if (which == 1) { kern = sk_b; skk = ss_b; bias = sb_b; xdim = kSE; }
    else if (which == 2) { kern = tk_f; skk = ts_f; bias = tb_f; xdim = kTE; }
    else                 { kern = tk_b; skk = ts_b; bias = tb_b; xdim = kTE; }

    if (which < 2) {
        if (j < kSE) xv[j] = song_emb[(size_t)songs[bt] * kSE + j];
    } else {
        if (j < kTE) xv[j] = ws[GI_OFF + (size_t)bt * kTE + j];
    }
    if (j < kPE) sv[j] = skip_emb[(size_t)skips[bt] * kPE + j];
    __syncthreads();

    float acc = bias[j] + bias[2 * k3U + j]; // ib + sb
    for (int d = 0; d < xdim; d++) acc += xv[d] * kern[(size_t)d * k3U + j];
    for (int d = 0; d < kPE;  d++) acc += sv[d] * skk[(size_t)d * k3U + j];

    ws[(size_t)which * A_STRIDE + (size_t)bt * k3U + j] = acc;
}

// ---------------------------------------------------------------------------
// Persistent GRU scan. grid = (16 batch tiles, 4 path/dir), block = 384 (12 waves).
// Each wave owns one 16-wide slice of the 192 mi columns; mi = h @ rec + rb via
// 16 chained v_wmma_f32_16x16x4_f32. Recurrent weights preloaded into registers.
// Next step's "a" slice is prefetched (global_prefetch_b8) to overlap the
// global-load latency with the WMMA chain -- the scan is the serial critical
// path of the whole model (100 dependent steps).
// ---------------------------------------------------------------------------
__global__ __launch_bounds__(384) void gru_scan_kernel(
    float* __restrict__ ws,
    const int* __restrict__ songs, const int* __restrict__ tags,
    const float* __restrict__ sr_f, const float* __restrict__ sb_f,
    const float* __restrict__ sr_b, const float* __restrict__ sb_b,
    const float* __restrict__ tr_f, const float* __restrict__ tb_f,
    const float* __restrict__ tr_b, const float* __restrict__ tb_b) {

    int tile = blockIdx.x;       // batch tile (16 rows)
    int pd   = blockIdx.y;       // 0=s_f 1=s_b 2=t_f 3=t_b
    int path = pd >> 1;
    int bwd  = pd & 1;

    const float* rec; const float* bias;
    if (pd == 0)      { rec = sr_f; bias = sb_f; }
    else if (pd == 1) { rec = sr_b; bias = sb_b; }
    else if (pd == 2) { rec = tr_f; bias = tb_f; }
    else              { rec = tr_b; bias = tb_b; }

    const float* A = ws + (size_t)pd * A_STRIDE;
    float* seq = ws + SEQ_OFF + (size_t)path * SEQ_STRIDE;

    __shared__ float h_s[16 * kU];    // hidden state (16 batch rows x 64)
    __shared__ float mi_s[16 * k3U];  // h @ rec + rb

    int wave = threadIdx.x >> 5;
    int lane = threadIdx.x & 31;
    int Nl = lane & 15;
    int kh = lane >> 4;
    int col = wave * 16 + Nl;         // 0..191
    int b0 = tile * 16;

    // preload this lane's recurrent-weight column pairs (loop-invariant)
    float regB[32];
#pragma unroll
    for (int kk = 0; kk < 16; kk++) {
        int kb = kk * 4 + kh * 2;
        regB[2 * kk]     = rec[(size_t)kb * k3U + col];
        regB[2 * kk + 1] = rec[(size_t)(kb + 1) * k3U + col];
    }
    float rbv = bias[k3U + col];      // rb

    for (int i = threadIdx.x; i < 16 * kU; i += 384) h_s[i] = 0.f;
    __syncthreads();

    for (int step = 0; step < kT; ++step) {
        int t = bwd ? (kT - 1 - step) : step;

        // prefetch next step's "a" slice (16 rows x 768B = 96 cachelines of 128B)
        if (step + 1 < kT && threadIdx.x < 96) {
            int tn = bwd ? (kT - 2 - step) : (step + 1);
            int row = threadIdx.x / 6;          // 0..15
            int seg = threadIdx.x % 6;          // 6 x 128B per row
            const float* p = A + ((size_t)(b0 + row) * kT + tn) * k3U + seg * 32;
            __builtin_prefetch(p, 0, 3);
        }

        // mi tile = h(16x64) @ rec(64, col-slice) + rb
        v8f acc;
#pragma unroll
        for (int v = 0; v < 8; v++) acc[v] = rbv;
#pragma unroll
        for (int kk = 0; kk < 16; kk++) {
            int kb = kk * 4 + kh * 2;
            v2f a, bb;
            a.x = h_s[Nl * kU + kb];
            a.y = h_s[Nl * kU + kb + 1];
            bb.x = regB[2 * kk];
            bb.y = regB[2 * kk + 1];
            acc = __builtin_amdgcn_wmma_f32_16x16x4_f32(
                false, a, false, bb, (short)0, acc, false, false);
        }
#pragma unroll
        for (int v = 0; v < 8; v++)
            mi_s[(v + kh * 8) * k3U + col] = acc[v];
        __syncthreads();

        // gate math + state update (1024 elements over 384 threads)
        for (int idx = threadIdx.x; idx < 16 * kU; idx += 384) {
            int m = idx >> 6, u = idx & 63;
            int b = b0 + m;
            const float* arow = A + ((size_t)b * kT + t) * k3U;
            float z  = hsig(arow[u]        + mi_s[m * k3U + u]);
            float r  = hsig(arow[kU + u]   + mi_s[m * k3U + kU + u]);
            float hh = tanhf(arow[2*kU + u] + r * mi_s[m * k3U + 2*kU + u]);
            float hold = h_s[idx];
            float hn = z * hold + (1.f - z) * hh;
            bool msk = (path == 0) ? (songs[(size_t)b * kT + t] != 0)
                                   : (tags[((size_t)b * kT + t) * kK] != 0);
            float hval = msk ? hn : hold;
            h_s[idx] = hval;
            seq[((size_t)b * kT + t) * 128 + bwd * kU + u] = hval;
        }
        __syncthreads();
    }

    if (!bwd) {
        float* hout = ws + H_OFF + (size_t)path * kB * kU;
        for (int idx = threadIdx.x; idx < 16 * kU; idx += 384)
            hout[(size_t)b0 * kU + idx] = h_s[idx];
    }
}

// ---------------------------------------------------------------------------
// Attention pooling. grid = (B, 2 paths), block = 128.
// ---------------------------------------------------------------------------
__global__ void attn_kernel(float* __restrict__ ws,
                            const int* __restrict__ songs, const int* __restrict__ tags,
                            const float* __restrict__ sWk, const float* __restrict__ sbk,
                            const float* __restrict__ sWq, const float* __restrict__ sbq,
                            const float* __restrict__ sWe, const float* __restrict__ sbe,
                            const float* __restrict__ tWk, const float* __restrict__ tbk,
                            const float* __restrict__ tWq, const float* __restrict__ tbq,
                            const float* __restrict__ tWe, const float* __restrict__ tbe) {
    int b = blockIdx.x, path = blockIdx.y, j = threadIdx.x;
    const float *Wk, *bk, *Wq, *bq, *We; float be0;
    if (path == 0) { Wk = sWk; bk = sbk; Wq = sWq; bq = sbq; We = sWe; be0 = sbe[0]; }
    else           { Wk = tWk; bk = tbk; Wq = tWq; bq = tbq; We = tWe; be0 = tbe[0]; }

    const float* seq = ws + SEQ_OFF + (size_t)path * SEQ_STRIDE + (size_t)b * kT * 128;
    const float* h   = ws + H_OFF + (size_t)path * kB * kU + (size_t)b * kU;

    __shared__ float qq[kU];
    __shared__ float e[kT];
    __shared__ float red[kU];
    __shared__ float mx_s, sum_s;

    if (j < kU) {
        float s = bq[j];
        for (int d = 0; d < kU; d++) s += h[d] * Wq[(size_t)d * kU + j];
        qq[j] = s;
    }
    __syncthreads();

    for (int t = 0; t < kT; t++) {
        if (j < kU) {
            float s = bk[j];
            const float* sr = seq + (size_t)t * 128;
            for (int d = 0; d < 128; d++) s += sr[d] * Wk[(size_t)d * kU + j];
            red[j] = tanhf(s + qq[j]) * We[j];
        }
        __syncthreads();
        for (int st = 32; st > 0; st >>= 1) {
            if (j < st) red[j] += red[j + st];
            __syncthreads();
        }
        if (j == 0) {
            bool msk = (path == 0) ? (songs[(size_t)b * kT + t] != 0)
                                   : (tags[((size_t)b * kT + t) * kK] != 0);
            e[t] = red[0] + be0 + (msk ? 0.f : -1.0e9f);
        }
        __syncthreads();
    }

    if (j == 0) {
        float m = -1.0e30f;
        for (int t = 0; t < kT; t++) m = fmaxf(m, e[t]);
        mx_s = m;
    }
    __syncthreads();
    if (j < kT) e[j] = expf(e[j] - mx_s);
    __syncthreads();
    if (j == 0) {
        float s = 0.f;
        for (int t = 0; t < kT; t++) s += e[t];
        sum_s = s;
    }
    __syncthreads();

    float c = 0.f;
    for (int t = 0; t < kT; t++) c += e[t] * seq[(size_t)t * 128 + j];
    ws[CTX_OFF + (size_t)b * 256 + path * 128 + j] = c / sum_s;
}

// ---------------------------------------------------------------------------
// v = relu(ctx @ vW + vb), zero-padded 50 -> 64
// ---------------------------------------------------------------------------
__global__ void v_kernel(float* __restrict__ ws,
                         const float* __restrict__ vW, const float* __restrict__ vb) {
    int b = blockIdx.x, j = threadIdx.x; // 64 threads
    float r = 0.f;
    if (j < kVL) {
        float s = vb[j];
        const float* c = ws + CTX_OFF + (size_t)b * 256;
        for (int d = 0; d < 256; d++) s += c[d] * vW[(size_t)d * kVL + j];
        r = fmaxf(s, 0.f);
    }
    ws[V_OFF + (size_t)b * kU + j] = r;
}

// ---------------------------------------------------------------------------
// out = v @ oW + ob, (256 x 100000), WMMA f32 16x16x4. One wave per 16x16 tile.
// K padded 50 -> 52 (v buffer has zeros at 50..63, oW rows clamped).
// ---------------------------------------------------------------------------
__global__ __launch_bounds__(256) void out_gemm_kernel(const float* __restrict__ ws,
                                                       const float* __restrict__ oW,
                                                       const float* __restrict__ ob,
                                                       float* __restrict__ out) {
    int wave = threadIdx.x >> 5, lane = threadIdx.x & 31;
    int Nl = lane & 15, kh = lane >> 4;
    int ntile = blockIdx.x * 8 + wave;
    if (ntile >= kSV / 16) return;  // wave-uniform
    int n0 = ntile * 16;
    int m0 = blockIdx.y * 16;
    const float* V = ws + V_OFF;

    v8f acc;
    float obv = ob[n0 + Nl];
#pragma unroll
    for (int v = 0; v < 8; v++) acc[v] = obv;

#pragma unroll
    for (int kk = 0; kk < 13; kk++) {
        int kb = kk * 4 + kh * 2;            // max 50 -> reads V[..50],V[..51] == 0
        v2f a, bb;
        a.x = V[(size_t)(m0 + Nl) * kU + kb];
        a.y = V[(size_t)(m0 + Nl) * kU + kb + 1];
        int r0 = (kb < kVL) ? kb : (kVL - 1);
        int r1 = (kb + 1 < kVL) ? (kb + 1) : (kVL - 1);
        bb.x = oW[(size_t)r0 * kSV + n0 + Nl];
        bb.y = oW[(size_t)r1 * kSV + n0 + Nl];
        acc = __builtin_amdgcn_wmma_f32_16x16x4_f32(
            false, a, false, bb, (short)0, acc, false, false);
    }
#pragma unroll
    for (int v = 0; v < 8; v++) {
        int m = m0 + v + kh * 8;
        out[(size_t)m * kSV + n0 + Nl] = acc[v];
    }
}

// ---------------------------------------------------------------------------
extern "C" void kernel_launch(void* const* d_in, const int* in_sizes, int n_in,
                              void* d_out, int out_size, void* d_ws, size_t ws_size,
                              hipStream_t stream) {
    const int* songs = (const int*)d_in[0];
    const int* tags  = (const int*)d_in[1];
    const int* skips = (const int*)d_in[2];
    const float* song_emb = (const float*)d_in[3];
    const float* tag_emb  = (const float*)d_in[4];
    const float* skip_emb = (const float*)d_in[5];
    const float* sk_f = (const float*)d_in[6];
    const float* sr_f = (const float*)d_in[7];
    const float* ss_f = (const float*)d_in[8];
    const float* sb_f = (const float*)d_in[9];
    const float* sk_b = (const float*)d_in[10];
    const float* sr_b = (const float*)d_in[11];
    const float* ss_b = (const float*)d_in[12];
    const float* sb_b = (const float*)d_in[13];
    const float* sWk = (const float*)d_in[14];
    const float* sbk = (const float*)d_in[15];
    const float* sWq = (const float*)d_in[16];
    const float* sbq = (const float*)d_in[17];
    const float* sWe = (const float*)d_in[18];
    const float* sbe = (const float*)d_in[19];
    const float* tk_f = (const float*)d_in[20];
    const float* tr_f = (const float*)d_in[21];
    const float* ts_f = (const float*)d_in[22];
    const float* tb_f = (const float*)d_in[23];
    const float* tk_b = (const float*)d_in[24];
    const float* tr_b = (const float*)d_in[25];
    const float* ts_b = (const float*)d_in[26];
    const float* tb_b = (const float*)d_in[27];
    const float* tWk = (const float*)d_in[28];
    const float* tbk = (const float*)d_in[29];
    const float* tWq = (const float*)d_in[30];
    const float* tbq = (const float*)d_in[31];
    const float* tWe = (const float*)d_in[32];
    const float* tbe = (const float*)d_in[33];
    const float* vW = (const float*)d_in[34];
    const float* vb = (const float*)d_in[35];
    const float* oW = (const float*)d_in[36];
    const float* ob = (const float*)d_in[37];

    float* ws = (float*)d_ws;
    float* out = (float*)d_out;

    gi_kernel<<<(kB * kT + 127) / 128, 128, 0, stream>>>(tags, tag_emb, ws);

    precompute_a_kernel<<<dim3(kB * kT, 4), k3U, 0, stream>>>(
        songs, skips, song_emb, skip_emb,
        sk_f, ss_f, sb_f, sk_b, ss_b, sb_b,
        tk_f, ts_f, tb_f, tk_b, ts_b, tb_b, ws);

    gru_scan_kernel<<<dim3(kB / 16, 4), 384, 0, stream>>>(
        ws, songs, tags,
        sr_f, sb_f, sr_b, sb_b, tr_f, tb_f, tr_b, tb_b);

    attn_kernel<<<dim3(kB, 2), 128, 0, stream>>>(
        ws, songs, tags,
        sWk, sbk, sWq, sbq, sWe, sbe,
        tWk, tbk, tWq, tbq, tWe, tbe);

    v_kernel<<<kB, kU, 0, stream>>>(ws, vW, vb);

    out_gemm_kernel<<<dim3((kSV / 16 + 7) / 8, kB / 16), 256, 0, stream>>>(
        ws, oW, ob, out);
}